// GPKANLayer_63934883168343
// MI455X (gfx1250) — compile-verified
//
#include <hip/hip_runtime.h>
#include <stdint.h>

// ---------------------------------------------------------------------------
// SVGP-KAN layer (RBF kernel, deterministic input path) for MI455X / gfx1250.
//
// Shapes: x(1024,64)  z,q_mu,q_log_var(64,64,20)  log_scale,log_variance(64,64)
// Out:    y_mean(1024,64) ++ y_var(1024,64)  (concatenated flat, fp32)
//
// Compute-bound on v_exp_f32 (~170M exps); memory is negligible (~1.7 MB total
// vs 23.3 TB/s). Strategy: fold all per-(o,i,m) constants once per block into
// LDS, run the hot loop as 2 trans + ~10 VALU per edge element with uniform-
// address LDS broadcast reads (block shares one `o`), 256-thread blocks (8 wave32).
// CDNA5-specific paths: async global->LDS staging + s_wait_asynccnt (guarded),
// global_prefetch_b8.
// ---------------------------------------------------------------------------

#ifndef __has_builtin
#define __has_builtin(x) 0
#endif

#if __has_builtin(__builtin_amdgcn_global_load_async_to_lds_b32) && \
    __has_builtin(__builtin_amdgcn_s_wait_asynccnt)
#define SVGP_ASYNC_LDS 1
#else
#define SVGP_ASYNC_LDS 0
#endif

#define SVGP_BATCH 1024
#define SVGP_INF   64
#define SVGP_OUTF  64
#define SVGP_M     20
#define SVGP_EDGE  (SVGP_INF * SVGP_M)   // 1280 elements per output slice
#define SVGP_LOG2E 1.4426950408889634f

// Builtin signature (from hipcc diagnostic): arg0 is a non-const AS(1) int*,
// arg1 an AS(3) int*, then imm offset + imm cpol.
typedef __attribute__((address_space(1))) int* svgp_g32_t;
typedef __attribute__((address_space(3))) int* svgp_l32_t;

__global__ __launch_bounds__(256)
void svgp_kan_kernel(const float* __restrict__ x,
                     const float* __restrict__ z,
                     const float* __restrict__ qmu,
                     const float* __restrict__ qlv,
                     const float* __restrict__ lsc,
                     const float* __restrict__ lvr,
                     float* __restrict__ out)
{
    // Parameter slice for this block's output feature `o` (~16.4 KB of LDS).
    __shared__ float ZL [SVGP_EDGE];   // z[o,i,m]
    __shared__ float W1L[SVGP_EDGE];   // c1(o,i) * q_mu
    __shared__ float W2L[SVGP_EDGE];   // c2(o,i) * (q_mu^2 + q_var)
    __shared__ float C1L[SVGP_INF], C2L[SVGP_INF];
    __shared__ float NK1L[SVGP_INF], NK2L[SVGP_INF];

    const int tid   = threadIdx.x;
    const int o     = blockIdx.x >> 2;         // 64 output features
    const int bbase = (blockIdx.x & 3) << 8;   // 4 chunks of 256 batches
    const int b     = bbase + tid;
    const int sbase = o * SVGP_EDGE;

    // ---- stage z[o,:,:] into LDS (CDNA5 async DMA path) ----
#if SVGP_ASYNC_LDS
    for (int idx = tid; idx < SVGP_EDGE; idx += 256) {
        __builtin_amdgcn_global_load_async_to_lds_b32(
            (svgp_g32_t)(uintptr_t)(z + sbase + idx),
            (svgp_l32_t)(uint32_t)(uintptr_t)&ZL[idx],
            /*offset=*/0, /*cpol=*/0);
    }
#endif

    // ---- per-(o,i) folded constants (64 lanes, done once per block) ----
    if (tid < SVGP_INF) {
        const int i = tid;
        float ell = fmaxf(__expf(lsc[o * SVGP_INF + i]), 0.1f);
        float s2  = fmaxf(__expf(lvr[o * SVGP_INF + i]), 1e-5f);
        float l2   = ell * ell;
        float den1 = l2 + 1e-6f;          // l^2 + x_var
        float den2 = l2 + 2e-6f;          // l^2 + 2*x_var
        C1L[i]  = s2 * sqrtf(l2 / den1);
        C2L[i]  = s2 * s2 * sqrtf(l2 / den2);
        NK1L[i] = -(0.5f * SVGP_LOG2E) / den1;  // exp(-d2/(2den1)) = exp2(d2*NK1)
        NK2L[i] = -SVGP_LOG2E / den2;           // exp(-d2/den2)    = exp2(d2*NK2)
    }
    __syncthreads();

    // ---- fold q_mu / q_var into W1/W2 (kills the per-batch-redundant exps) --
    for (int idx = tid; idx < SVGP_EDGE; idx += 256) {
        const int i = idx / SVGP_M;
        float qm = qmu[sbase + idx];
        float qv = fmaxf(__expf(qlv[sbase + idx]), 1e-5f);
        W1L[idx] = C1L[i] * qm;
        W2L[idx] = C2L[i] * __builtin_fmaf(qm, qm, qv);
#if !SVGP_ASYNC_LDS
        ZL[idx] = z[sbase + idx];
#endif
    }
#if SVGP_ASYNC_LDS
    __builtin_amdgcn_s_wait_asynccnt(0);   // our wave's async->LDS done
#endif
    __syncthreads();                       // all staging visible to all waves

    // ---- hot loop: one (b,o) per thread; params broadcast from LDS ----
    const float* __restrict__ xrow = x + (size_t)b * SVGP_INF;
    __builtin_prefetch(xrow, 0, 0);        // global_prefetch_b8 on gfx1250

    float acc_mean = 0.0f, acc_var = 0.0f;
    for (int i4 = 0; i4 < SVGP_INF; i4 += 4) {
        const float4 x4 = *(const float4*)(xrow + i4);  // global_load_b128
        #pragma unroll
        for (int ii = 0; ii < 4; ++ii) {
            const int   i   = i4 + ii;
            const float xb  = (ii == 0) ? x4.x : (ii == 1) ? x4.y
                            : (ii == 2) ? x4.z : x4.w;
            const float nk1 = NK1L[i];
            const float nk2 = NK2L[i];
            float var_i = 0.0f;
            #pragma unroll
            for (int m = 0; m < SVGP_M; ++m) {
                const int   idx = i * SVGP_M + m;
                const float d   = xb - ZL[idx];
                const float d2  = d * d;
                const float e1  = __builtin_amdgcn_exp2f(d2 * nk1);  // v_exp_f32
                const float e2  = __builtin_amdgcn_exp2f(d2 * nk2);  // v_exp_f32
                const float t   = e1 * W1L[idx];                     // psi1*qmu
                acc_mean += t;
                var_i = __builtin_fmaf(e2, W2L[idx], var_i);  // + ek2*(qm^2+qv)
                var_i = __builtin_fmaf(-t, t, var_i);         // - psi1^2*qm^2
            }
            acc_var += fmaxf(var_i, 0.0f);   // per-edge clip BEFORE i-sum
        }
    }

    out[(size_t)b * SVGP_OUTF + o] = acc_mean;
    out[(size_t)SVGP_BATCH * SVGP_OUTF + (size_t)b * SVGP_OUTF + o] = acc_var;
}

extern "C" void kernel_launch(void* const* d_in, const int* in_sizes, int n_in,
                              void* d_out, int out_size, void* d_ws, size_t ws_size,
                              hipStream_t stream) {
    (void)in_sizes; (void)n_in; (void)out_size; (void)d_ws; (void)ws_size;
    const float* x   = (const float*)d_in[0];
    const float* z   = (const float*)d_in[1];
    const float* qmu = (const float*)d_in[2];
    const float* qlv = (const float*)d_in[3];
    const float* lsc = (const float*)d_in[4];
    const float* lvr = (const float*)d_in[5];
    float* out = (float*)d_out;

    // grid: 64 output features x 4 batch chunks; block: 256 threads (8 wave32)
    svgp_kan_kernel<<<dim3(SVGP_OUTF * 4), dim3(256), 0, stream>>>(
        x, z, qmu, qlv, lsc, lvr, out);
}